// Decomposition4D_52372831207553
// MI455X (gfx1250) — compile-verified
//
#include <hip/hip_runtime.h>
#include <cstdint>

#define TBLSZ   (1u << 19)
#define TMASK   (TBLSZ - 1u)
#define PRIME1  2654435761u
#define PRIME2  805459861u
#define NLEV    16
#define BLOCK   256
#define LDS_STRIDE 36   // 36 words: 144B row stride -> 16B-aligned chunks, <=2-way DS conflicts

// scales[l] = float(32 * (2048/32)^(l/15) - 1) = 32*2^(0.4 l) - 1  (matches reference doubles)
__constant__ float kScales[NLEV] = {
    31.0f,            41.22425314473261f, 54.71523605095194f, 72.51669471981023f,
    96.00586025666548f, 127.0f,           167.89701257893046f, 221.86094420380777f,
    293.06677887924093f, 387.02344102666194f, 511.0f,          674.5880503157218f,
    890.4437768152311f, 1175.2671155169637f, 1551.0937641066478f, 2047.0f};

__device__ __forceinline__ void interp1d_setup(float c, int& i0, float& w) {
    float pos = c * 2047.0f;
    float f   = floorf(pos);
    f  = fminf(fmaxf(f, 0.0f), 2046.0f);
    i0 = (int)f;
    w  = pos - f;
}

// 8-corner trilinear hash gather for one (plane, level).
// a*,b*,c* are the per-dim corner hashes, already multiplied by their dim's prime.
__device__ __forceinline__ void gather8(const float2* __restrict__ tb, uint32_t lbase,
                                        uint32_t a0, uint32_t a1,
                                        uint32_t b0, uint32_t b1,
                                        uint32_t c0, uint32_t c1,
                                        float wa, float wb, float wc,
                                        float& f0, float& f1) {
    const uint32_t A[2] = {a0, a1};
    const uint32_t B[2] = {b0, b1};
    const uint32_t C[2] = {c0, c1};
    const float   WA[2] = {1.0f - wa, wa};
    const float   WB[2] = {1.0f - wb, wb};
    const float   WC[2] = {1.0f - wc, wc};
    float s0 = 0.0f, s1 = 0.0f;
#pragma unroll
    for (int c = 0; c < 8; ++c) {
        uint32_t idx = (A[c & 1] ^ B[(c >> 1) & 1] ^ C[(c >> 2) & 1]) & TMASK;
        float2 g = tb[lbase + idx];                       // global_load_b64 gather
        float  w = WA[c & 1] * WB[(c >> 1) & 1] * WC[(c >> 2) & 1];
        s0 = fmaf(w, g.x, s0);
        s1 = fmaf(w, g.y, s1);
    }
    f0 = s0; f1 = s1;
}

__global__ __launch_bounds__(BLOCK)
void dec4d_kernel(const float* __restrict__ xyz, const float* __restrict__ times,
                  const float* __restrict__ tab_xyz, const float* __restrict__ tab_xyt,
                  const float* __restrict__ tab_yzt, const float* __restrict__ tab_xzt,
                  const float* __restrict__ vecs, float* __restrict__ out) {
    __shared__ float lds[BLOCK * LDS_STRIDE];            // 36 KB

    const int tid = threadIdx.x;
    const int n   = blockIdx.x * BLOCK + tid;

    const float x = xyz[3 * n + 0];
    const float y = xyz[3 * n + 1];
    const float z = xyz[3 * n + 2];
    const float t = times[n];

    // interp1d setups: vectors[0]@t, vectors[1]@z, vectors[2]@x, vectors[3]@y
    int   vi0, vi1, vi2, vi3;
    float vw0, vw1, vw2, vw3;
    interp1d_setup(t, vi0, vw0);
    interp1d_setup(z, vi1, vw1);
    interp1d_setup(x, vi2, vw2);
    interp1d_setup(y, vi3, vw3);

    const float2* vr0 = (const float2*)(vecs);
    const float2* vr1 = (const float2*)(vecs + 65536);
    const float2* vr2 = (const float2*)(vecs + 131072);
    const float2* vr3 = (const float2*)(vecs + 196608);

    const float2* tb0 = (const float2*)tab_xyz;
    const float2* tb1 = (const float2*)tab_xyt;
    const float2* tb2 = (const float2*)tab_yzt;
    const float2* tb3 = (const float2*)tab_xzt;

#pragma unroll 2
    for (int l = 0; l < NLEV; ++l) {
        const float s = kScales[l];

        // Per-dimension quantization (shared across the 4 planes).
        float px = fmaf(x, s, 0.5f), fx = floorf(px), wx = px - fx;
        float py = fmaf(y, s, 0.5f), fy = floorf(py), wy = py - fy;
        float pz = fmaf(z, s, 0.5f), fz = floorf(pz), wz = pz - fz;
        float pt = fmaf(t, s, 0.5f), ft = floorf(pt), wt = pt - ft;
        uint32_t cx = (uint32_t)fx, cy = (uint32_t)fy;
        uint32_t cz = (uint32_t)fz, ct = (uint32_t)ft;

        // Pre-multiplied per-dim corner hashes (dim0 prime == 1).
        uint32_t X0  = cx,                 X1  = cx + 1u;
        uint32_t Y0  = cy,                 Y1  = cy + 1u;
        uint32_t Y0p = cy * PRIME1,        Y1p = (cy + 1u) * PRIME1;
        uint32_t Z0p1 = cz * PRIME1,       Z1p1 = (cz + 1u) * PRIME1;
        uint32_t Z0p2 = cz * PRIME2,       Z1p2 = (cz + 1u) * PRIME2;
        uint32_t T0p2 = ct * PRIME2,       T1p2 = (ct + 1u) * PRIME2;

        const uint32_t lbase = (uint32_t)l * TBLSZ;

        float f00, f01, f10, f11, f20, f21, f30, f31;
        gather8(tb0, lbase, X0, X1, Y0p, Y1p, Z0p2, Z1p2, wx, wy, wz, f00, f01); // (x,y,z)
        gather8(tb1, lbase, X0, X1, Y0p, Y1p, T0p2, T1p2, wx, wy, wt, f10, f11); // (x,y,t)
        gather8(tb2, lbase, Y0, Y1, Z0p1, Z1p1, T0p2, T1p2, wy, wz, wt, f20, f21); // (y,z,t)
        gather8(tb3, lbase, X0, X1, Z0p1, Z1p1, T0p2, T1p2, wx, wz, wt, f30, f31); // (x,z,t)

        // Per-level columns (2l, 2l+1) of the four interp1d results (L2-resident).
        float2 a, b;
        a = vr0[vi0 * 16 + l]; b = vr0[vi0 * 16 + 16 + l];
        float v00 = fmaf(vw0, b.x - a.x, a.x), v01 = fmaf(vw0, b.y - a.y, a.y);
        a = vr1[vi1 * 16 + l]; b = vr1[vi1 * 16 + 16 + l];
        float v10 = fmaf(vw1, b.x - a.x, a.x), v11 = fmaf(vw1, b.y - a.y, a.y);
        a = vr2[vi2 * 16 + l]; b = vr2[vi2 * 16 + 16 + l];
        float v20 = fmaf(vw2, b.x - a.x, a.x), v21 = fmaf(vw2, b.y - a.y, a.y);
        a = vr3[vi3 * 16 + l]; b = vr3[vi3 * 16 + 16 + l];
        float v30 = fmaf(vw3, b.x - a.x, a.x), v31 = fmaf(vw3, b.y - a.y, a.y);

        float a0 = fmaf(f00, v00, fmaf(f10, v10, fmaf(f20, v20, f30 * v30)));
        float a1 = fmaf(f01, v01, fmaf(f11, v11, fmaf(f21, v21, f31 * v31)));

        // Stage into LDS tile (ds_store_b64; 144B row stride keeps banks ~clean).
        *(float2*)&lds[tid * LDS_STRIDE + 2 * l] = make_float2(a0, a1);
    }

    __syncthreads();

    // Epilogue: async LDS -> global stores (CDNA5 path, tracked by ASYNCcnt).
    // Block output = 256 points * 32 floats = 32KB = 2048 16B chunks.
    // IMPORTANT: derive the LDS byte address from the actual shared pointer.
    // The ptrtoint captures `lds`, so the "memory"-clobbering asm is a visible
    // reader and the LDS stores (and the whole compute) cannot be DCE'd.
    // Generic LDS pointer layout is {shared_aperture, lds_offset[31:0]}, so the
    // low 32 bits are exactly the VSRC byte address the instruction needs.
    float* outp = out + (size_t)blockIdx.x * (BLOCK * 32);
    for (int i = tid; i < BLOCK * 8; i += BLOCK) {
        const int p = i >> 3;          // point in block
        const int q = i & 7;           // 16B chunk within point
        uint32_t la = (uint32_t)(uintptr_t)&lds[p * LDS_STRIDE + q * 4];
        uint64_t ga = (uint64_t)(uintptr_t)(outp + (size_t)(p * 32 + q * 4));
        asm volatile("global_store_async_from_lds_b128 %0, %1, off"
                     :: "v"(ga), "v"(la) : "memory");
    }
    asm volatile("s_wait_asynccnt 0" ::: "memory");
}

extern "C" void kernel_launch(void* const* d_in, const int* in_sizes, int n_in,
                              void* d_out, int out_size, void* d_ws, size_t ws_size,
                              hipStream_t stream) {
    const float* xyz     = (const float*)d_in[0];
    const float* times   = (const float*)d_in[1];
    const float* tab_xyz = (const float*)d_in[2];
    const float* tab_xyt = (const float*)d_in[3];
    const float* tab_yzt = (const float*)d_in[4];
    const float* tab_xzt = (const float*)d_in[5];
    const float* vecs    = (const float*)d_in[6];
    float* out = (float*)d_out;

    const int N = in_sizes[0] / 3;          // 524288
    dim3 grid(N / BLOCK), block(BLOCK);
    hipLaunchKernelGGL(dec4d_kernel, grid, block, 0, stream,
                       xyz, times, tab_xyz, tab_xyt, tab_yzt, tab_xzt, vecs, out);
}